// Attention_30004641530540
// MI455X (gfx1250) — compile-verified
//
#include <hip/hip_runtime.h>

typedef float v2f __attribute__((ext_vector_type(2)));
typedef float v8f __attribute__((ext_vector_type(8)));

#if __has_builtin(__builtin_amdgcn_tensor_load_to_lds)
#define HAVE_TDM 1
#else
#define HAVE_TDM 0
#endif

#define B_N      4
#define C_N      256          // channels == head dim
#define N_N      4096         // 64*64 spatial positions
#define M_TILE   64           // query rows per block (16 per wave)
#define K_TILE   64           // keys per iteration
#define NTILES   (N_N / K_TILE)
#define NWAVES   4
#define NTHREADS 128
#define LSTRIDE  68           // padded LDS row stride (floats): 16B-aligned, bank-friendly
#define QS_F     (C_N * LSTRIDE)           // 17408 floats
#define KS_F     (C_N * LSTRIDE)           // 17408 floats (per K buffer)
#define PS_F     (NWAVES * 16 * LSTRIDE)   // 4352 floats
#define SMEM_FLOATS (QS_F + 2 * KS_F + PS_F)  // 56576 floats = 226304 B (< 320KB WGP LDS)

#if HAVE_TDM
typedef unsigned int tdm_v4u __attribute__((ext_vector_type(4)));
typedef int          tdm_v4i __attribute__((ext_vector_type(4)));
typedef int          tdm_v8i __attribute__((ext_vector_type(8)));

// DMA a [C_N rows x K_TILE cols] f32 tile (row stride N_N floats in memory) into LDS,
// laid out [dim][pos] with LSTRIDE=68 float rows via TDM pad (64 dwords + 4 dwords pad).
__device__ __forceinline__ void tdm_load_tile(const float* gsrc, unsigned lds_byte_off)
{
    unsigned long long ga = (unsigned long long)(uintptr_t)gsrc;
    tdm_v4u g0;
    g0[0] = 1u;                                               // count=1, user descriptor
    g0[1] = lds_byte_off;                                     // lds_addr (bytes)
    g0[2] = (unsigned)ga;                                     // global_addr[31:0]
    g0[3] = (unsigned)((ga >> 32) & 0x01FFFFFFull)            // global_addr[56:32]
          | (2u << 30);                                       // type = 2 ("image")
    tdm_v8i g1;
    g1[0] = (2 << 16)      // data_size = 4 bytes
          | (1 << 20)      // pad_enable
          | (5 << 22)      // pad_interval: 64 dwords
          | (3 << 25);     // pad_amount: 4 dwords  -> LDS row stride 68 floats
    g1[1] = (int)(((unsigned)N_N & 0xFFFFu) << 16);           // tensor_dim0[15:0] = 4096
    g1[2] = (int)((((unsigned)N_N >> 16) & 0xFFFFu)           // tensor_dim0[31:16]
          | ((unsigned)C_N << 16));                           // tensor_dim1[15:0] = 256
    g1[3] = (int)(((unsigned)C_N >> 16)                       // tensor_dim1[31:16]
          | ((unsigned)K_TILE << 16));                        // tile_dim0 = 64
    g1[4] = (int)C_N;                                         // tile_dim1 = 256, tile_dim2 = 0
    g1[5] = (int)N_N;                                         // tensor_dim0_stride[31:0] = 4096
    g1[6] = 0;                                                // stride hi / dim1_stride (2D: unused)
    g1[7] = 0;
    tdm_v4i gz = {0, 0, 0, 0};
#if defined(__clang_major__) && (__clang_major__ >= 23)
    tdm_v8i gz8 = {0, 0, 0, 0, 0, 0, 0, 0};
    __builtin_amdgcn_tensor_load_to_lds(g0, g1, gz, gz, gz8, 0);
#else
    __builtin_amdgcn_tensor_load_to_lds(g0, g1, gz, gz, 0);
#endif
}
#endif // HAVE_TDM

__global__ __launch_bounds__(NTHREADS)
void attn_flash_f32_wmma(const float* __restrict__ x, float* __restrict__ out)
{
    extern __shared__ float smem[];
    float* Qs  = smem;                        // [dim][row], stride LSTRIDE
    float* Kb0 = smem + QS_F;                 // K buffer 0: [dim][key], stride LSTRIDE
    float* Kb1 = smem + QS_F + KS_F;          // K buffer 1
    float* Ps  = smem + QS_F + 2 * KS_F;      // per-wave P staging

    const int b     = blockIdx.y;
    const int qbase = blockIdx.x * M_TILE;
    const int tid   = threadIdx.x;
    const int wave  = tid >> 5;
    const int lane  = tid & 31;
    const int l16   = lane & 15;
    const int hi    = lane >> 4;      // which 16-lane half
    const int k0    = hi ? 2 : 0;     // K-offsets for A/B fragment per ISA 16x4 layout
    const int k1    = k0 + 1;

    const float* xb = x + (size_t)b * C_N * N_N;

#if HAVE_TDM
    const unsigned lds_base = (unsigned)(uintptr_t)(void*)smem;
    const unsigned koff[2]  = { lds_base + QS_F * 4u, lds_base + (QS_F + KS_F) * 4u };
    if (wave == 0) {
        tdm_load_tile(xb + qbase, lds_base);          // Q tile
        tdm_load_tile(xb + 0,     koff[0]);           // K tile 0
        __builtin_amdgcn_s_wait_tensorcnt(0);
    }
    __syncthreads();
#else
    // ---- fallback: cooperative Q tile load, float4
    #pragma unroll
    for (int it = 0; it < (M_TILE * C_N) / (4 * NTHREADS); ++it) {
        int lin  = it * NTHREADS + tid;
        int row4 = (lin & (M_TILE / 4 - 1)) * 4;
        int dim  = lin / (M_TILE / 4);
        const float4 q = *(const float4*)(xb + (size_t)dim * N_N + qbase + row4);
        *(float4*)(Qs + dim * LSTRIDE + row4) = q;
    }
    __syncthreads();
#endif

    // ---- flash-attention running state (16 rows per wave)
    v8f O[16];
    #pragma unroll
    for (int t = 0; t < 16; ++t) O[t] = {};
    float mrow[8], lrow[8];
    #pragma unroll
    for (int r = 0; r < 8; ++r) { mrow[r] = -__builtin_inff(); lrow[r] = 0.0f; }

    const int qrow = wave * 16 + l16;
    float*    Pw   = Ps + wave * (16 * LSTRIDE);

    for (int kt = 0; kt < NTILES; ++kt) {
#if HAVE_TDM
        float* Kc = (kt & 1) ? Kb1 : Kb0;
        if (wave == 0 && kt + 1 < NTILES)             // overlap next tile DMA with compute
            tdm_load_tile(xb + (kt + 1) * K_TILE, koff[(kt + 1) & 1]);
#else
        float* Kc = Kb0;
        const int kbase = kt * K_TILE;
        __syncthreads();
        #pragma unroll
        for (int it = 0; it < (K_TILE * C_N) / (4 * NTHREADS); ++it) {
            int lin  = it * NTHREADS + tid;
            int key4 = (lin & (K_TILE / 4 - 1)) * 4;
            int dim  = lin / (K_TILE / 4);
            const float4 kv = *(const float4*)(xb + (size_t)dim * N_N + kbase + key4);
            *(float4*)(Kc + dim * LSTRIDE + key4) = kv;
        }
        __syncthreads();
#endif

        // ---- S = Q K^T : 16 rows x 64 keys, reduce over d=256 in K=4 steps
        v8f S[4];
        #pragma unroll
        for (int ct = 0; ct < 4; ++ct) S[ct] = {};

        #pragma unroll 8
        for (int ks = 0; ks < C_N / 4; ++ks) {
            const int kd = ks * 4;
            v2f a;
            a.x = Qs[(kd + k0) * LSTRIDE + qrow];
            a.y = Qs[(kd + k1) * LSTRIDE + qrow];
            #pragma unroll
            for (int ct = 0; ct < 4; ++ct) {
                const int key = ct * 16 + l16;
                v2f bf;
                bf.x = Kc[(kd + k0) * LSTRIDE + key];
                bf.y = Kc[(kd + k1) * LSTRIDE + key];
                S[ct] = __builtin_amdgcn_wmma_f32_16x16x4_f32(
                    false, a, false, bf, (short)0, S[ct], false, false);
            }
        }

        // ---- online softmax over this key tile (rows split across lane halves)
        float scale[8];
        #pragma unroll
        for (int r = 0; r < 8; ++r) {
            float mx = fmaxf(fmaxf(S[0][r], S[1][r]), fmaxf(S[2][r], S[3][r]));
            mx = fmaxf(mx, __shfl_xor(mx, 1));
            mx = fmaxf(mx, __shfl_xor(mx, 2));
            mx = fmaxf(mx, __shfl_xor(mx, 4));
            mx = fmaxf(mx, __shfl_xor(mx, 8));
            const float nm = fmaxf(mrow[r], mx);
            scale[r] = __expf(mrow[r] - nm);
            mrow[r]  = nm;
            float rs = 0.0f;
            #pragma unroll
            for (int ct = 0; ct < 4; ++ct) {
                const float p = __expf(S[ct][r] - nm);
                S[ct][r] = p;
                rs += p;
            }
            rs += __shfl_xor(rs, 1);
            rs += __shfl_xor(rs, 2);
            rs += __shfl_xor(rs, 4);
            rs += __shfl_xor(rs, 8);
            lrow[r] = lrow[r] * scale[r] + rs;
        }

        // rescale running output accumulator
        #pragma unroll
        for (int dt = 0; dt < 16; ++dt)
            #pragma unroll
            for (int r = 0; r < 8; ++r)
                O[dt][r] *= scale[r];

        // ---- stash P (C-layout -> row-major LDS, wave-private)
        #pragma unroll
        for (int ct = 0; ct < 4; ++ct)
            #pragma unroll
            for (int r = 0; r < 8; ++r)
                Pw[(r + hi * 8) * LSTRIDE + ct * 16 + l16] = S[ct][r];

        // ---- O += P @ K-tile  (A frags hoisted; contiguous v2f -> ds_load_b64)
        v2f afr[16];
        #pragma unroll
        for (int kk = 0; kk < 16; ++kk)
            afr[kk] = *(const v2f*)(Pw + l16 * LSTRIDE + kk * 4 + k0);
        #pragma unroll
        for (int dt = 0; dt < 16; ++dt) {
            const int dim = dt * 16 + l16;
            #pragma unroll
            for (int kk = 0; kk < 16; ++kk) {
                const v2f bf = *(const v2f*)(Kc + dim * LSTRIDE + kk * 4 + k0);
                O[dt] = __builtin_amdgcn_wmma_f32_16x16x4_f32(
                    false, afr[kk], false, bf, (short)0, O[dt], false, false);
            }
        }

#if HAVE_TDM
        if (wave == 0)
            __builtin_amdgcn_s_wait_tensorcnt(0);     // next tile resident before release
        __syncthreads();
#endif
    }

    // ---- epilogue: normalize, stage transposed in LDS, coalesced float4 stores
    __syncthreads();           // everyone done reading Qs before reuse as staging
    float* Os = Qs;            // [dim][row], stride LSTRIDE (fits: C_N*LSTRIDE)
    #pragma unroll
    for (int r = 0; r < 8; ++r) {
        const float inv = 1.0f / lrow[r];
        #pragma unroll
        for (int dt = 0; dt < 16; ++dt)
            Os[(dt * 16 + l16) * LSTRIDE + wave * 16 + hi * 8 + r] = O[dt][r] * inv;
    }
    __syncthreads();
    float* ob = out + (size_t)b * C_N * N_N;
    #pragma unroll
    for (int it = 0; it < (M_TILE * C_N) / (4 * NTHREADS); ++it) {
        int lin  = it * NTHREADS + tid;
        int row4 = (lin & (M_TILE / 4 - 1)) * 4;
        int dim  = lin / (M_TILE / 4);
        *(float4*)(ob + (size_t)dim * N_N + qbase + row4) =
            *(const float4*)(Os + dim * LSTRIDE + row4);
    }
}

extern "C" void kernel_launch(void* const* d_in, const int* in_sizes, int n_in,
                              void* d_out, int out_size, void* d_ws, size_t ws_size,
                              hipStream_t stream) {
    (void)in_sizes; (void)n_in; (void)d_ws; (void)ws_size; (void)out_size;
    const float* x  = (const float*)d_in[0];
    float*      out = (float*)d_out;
    dim3 grid(N_N / M_TILE, B_N);   // 64 query tiles x 4 batches = 256 workgroups
    dim3 block(NTHREADS);
    attn_flash_f32_wmma<<<grid, block, SMEM_FLOATS * sizeof(float), stream>>>(x, out);
}